// yolov8_target_22084721836339
// MI455X (gfx1250) — compile-verified
//
#include <hip/hip_runtime.h>
#include <stdint.h>

#ifndef __has_builtin
#define __has_builtin(x) 0
#endif

#if __has_builtin(__builtin_amdgcn_global_load_async_to_lds_b128) && \
    __has_builtin(__builtin_amdgcn_s_wait_asynccnt)
#define USE_ASYNC_LDS 1
#else
#define USE_ASYNC_LDS 0
#endif

#if __has_builtin(__builtin_amdgcn_s_wait_dscnt)
#define WAIT_DS0() __builtin_amdgcn_s_wait_dscnt(0)
#else
#define WAIT_DS0() asm volatile("s_wait_dscnt 0" ::: "memory")
#endif

#define N_ANCH 336000          // anchors
#define N_V4   (N_ANCH / 4)    // 84000 float4 per row
#define C_TOT  84              // 4 box rows + 80 class rows
#define K_SEL  6720            // int(N * 0.02)
#define CONF_T 0.2f
#define NB_SUM 512             // partial-sum blocks
#define EQ_CAP 2048
#define ROW_BLOCKS 4           // row-split factor for the streaming pass
#define ROWS_PER   20          // 80 class rows / ROW_BLOCKS
#define NBUF       8           // async pipeline depth (8 x 4KB LDS)

typedef int v4i __attribute__((vector_size(4 * sizeof(int))));
typedef __attribute__((address_space(1))) v4i* gptr_v4i;
typedef __attribute__((address_space(3))) v4i* lptr_v4i;

#if USE_ASYNC_LDS
#define ASYNC_LOAD16(g, l) \
  __builtin_amdgcn_global_load_async_to_lds_b128((gptr_v4i)(g), (lptr_v4i)(l), 0, 0)
#endif

struct SelState { unsigned prefix; unsigned mask; unsigned k_rem; unsigned pad; };

// Monotonic float<->uint maps: descending float order == descending uint order.
__device__ __forceinline__ unsigned f2u(float f) {
  unsigned u = __float_as_uint(f);
  return (u & 0x80000000u) ? ~u : (u | 0x80000000u);
}
__device__ __forceinline__ float u2f(unsigned u) {
  return __uint_as_float((u & 0x80000000u) ? (u & 0x7FFFFFFFu) : ~u);
}

__device__ __forceinline__ float4 fmax4(float4 a, float4 b) {
  return make_float4(fmaxf(a.x, b.x), fmaxf(a.y, b.y),
                     fmaxf(a.z, b.z), fmaxf(a.w, b.w));
}

// ---------------------------------------------------------------- init
__global__ void k_init(unsigned* __restrict__ hist, SelState* __restrict__ st,
                       float* __restrict__ eq_sum) {
  int t = threadIdx.x;
  if (t < 256) hist[t] = 0u;
  if (t == 0) { st->prefix = 0u; st->mask = 0u; st->k_rem = K_SEL; *eq_sum = 0.0f; }
}

__global__ void k_zero(unsigned* __restrict__ scores_u) {
  int i = blockIdx.x * 256 + threadIdx.x;
  if (i < N_ANCH) scores_u[i] = 0u;    // < f2u of any real float
}

// ---------------------------------------------------------------- scores
// scores_u[i] = sortable-uint of max over rows 4..83 of model[c*N + i].
// grid = (anchor tiles, ROW_BLOCKS).  Each block streams its 20 rows through
// an 8-deep async-to-LDS pipeline (asynccnt <= 7 steady state), then merges
// row-split partials with exact integer atomicMax (order-independent).
__global__ void __launch_bounds__(256)
k_scores(const float* __restrict__ model, unsigned* __restrict__ scores_u) {
  const int t   = threadIdx.x;
  const int v4  = blockIdx.x * 256 + t;
  const int v4c = (v4 < N_V4) ? v4 : (N_V4 - 1);   // clamp tail: uniform issue
  const int cbase = 4 + (int)blockIdx.y * ROWS_PER;
  const float4* m4 = (const float4*)model;

  float4 mx = make_float4(-3.402823466e38f, -3.402823466e38f,
                          -3.402823466e38f, -3.402823466e38f);

#if USE_ASYNC_LDS
  __shared__ __align__(16) float buf[NBUF][1024];  // 32 KB pipeline
  // prologue: fill all 8 stages
#pragma unroll
  for (int j = 0; j < NBUF; ++j)
    ASYNC_LOAD16(m4 + (size_t)(cbase + j) * N_V4 + v4c, &buf[j][t * 4]);
  // steady state: consume row k, refill slot with row k+NBUF
  for (int k = 0; k < ROWS_PER - NBUF; ++k) {
    __builtin_amdgcn_s_wait_asynccnt(NBUF - 1);    // oldest row landed
    float4 v = *(const float4*)&buf[k & (NBUF - 1)][t * 4];
    mx = fmax4(mx, v);
    WAIT_DS0();                                    // LDS read done before reuse
    ASYNC_LOAD16(m4 + (size_t)(cbase + k + NBUF) * N_V4 + v4c,
                 &buf[k & (NBUF - 1)][t * 4]);
  }
  // epilogue: drain remaining 8 rows
  __builtin_amdgcn_s_wait_asynccnt(0);
#pragma unroll
  for (int k = ROWS_PER - NBUF; k < ROWS_PER; ++k) {
    float4 v = *(const float4*)&buf[k & (NBUF - 1)][t * 4];
    mx = fmax4(mx, v);
  }
#else
  for (int c = cbase; c < cbase + ROWS_PER; ++c) {
    if (c + 2 < cbase + ROWS_PER)
      __builtin_prefetch(m4 + (size_t)(c + 2) * N_V4 + v4c, 0, 1);
    float4 v = m4[(size_t)c * N_V4 + v4c];
    mx = fmax4(mx, v);
  }
#endif

  if (v4 < N_V4) {
    atomicMax(&scores_u[v4 * 4 + 0], f2u(mx.x));
    atomicMax(&scores_u[v4 * 4 + 1], f2u(mx.y));
    atomicMax(&scores_u[v4 * 4 + 2], f2u(mx.z));
    atomicMax(&scores_u[v4 * 4 + 3], f2u(mx.w));
  }
}

// ---------------------------------------------------------------- histogram
// 256-bin histogram of one byte of scores_u, restricted to current prefix.
// LDS-privatized to avoid serializing hot-bin global atomics.
__global__ void __launch_bounds__(256)
k_hist(const unsigned* __restrict__ scores_u, unsigned* __restrict__ hist,
       const SelState* __restrict__ st, int shift) {
  __shared__ unsigned lh[256];
  lh[threadIdx.x] = 0u;
  __syncthreads();
  const unsigned prefix = st->prefix, mask = st->mask;
  for (int i = blockIdx.x * 256 + threadIdx.x; i < N_ANCH; i += gridDim.x * 256) {
    unsigned u = scores_u[i];
    if ((u & mask) == prefix) atomicAdd(&lh[(u >> shift) & 0xFFu], 1u);
  }
  __syncthreads();
  unsigned v = lh[threadIdx.x];
  if (v) atomicAdd(&hist[threadIdx.x], v);
}

// ---------------------------------------------------------------- select
__global__ void k_select(unsigned* __restrict__ hist, SelState* __restrict__ st,
                         int shift) {
  unsigned k = st->k_rem;
  int b = 255;
  for (; b > 0; --b) {
    unsigned h = hist[b];
    if (k <= h) break;
    k -= h;
  }
  st->prefix |= ((unsigned)b) << shift;
  st->mask   |= 0xFFu << shift;
  st->k_rem   = k;
  for (int i = 0; i < 256; ++i) hist[i] = 0u;      // ready for next pass
}

// ---------------------------------------------------------------- main sum
// Deterministic: fixed grid-stride order, LDS tree, per-block partials.
__global__ void __launch_bounds__(256)
k_sum(const float* __restrict__ model, const unsigned* __restrict__ scores_u,
      const SelState* __restrict__ st, float* __restrict__ partials) {
  const unsigned Tu = st->prefix;                  // bit pattern of K-th largest
  const unsigned CONF_U = f2u(CONF_T);
  float acc = 0.0f;
  for (int i = blockIdx.x * 256 + threadIdx.x; i < N_ANCH; i += gridDim.x * 256) {
    unsigned u = scores_u[i];
    if (u > Tu && u >= CONF_U) {
      acc += u2f(u) + model[0 * N_ANCH + i] + model[1 * N_ANCH + i]
                    + model[2 * N_ANCH + i] + model[3 * N_ANCH + i];
    }
  }
  __shared__ float red[256];
  red[threadIdx.x] = acc;
  __syncthreads();
  for (int s = 128; s > 0; s >>= 1) {
    if (threadIdx.x < s) red[threadIdx.x] += red[threadIdx.x + s];
    __syncthreads();
  }
  if (threadIdx.x == 0) partials[blockIdx.x] = red[0];
}

// ---------------------------------------------------------------- boundary ties
// Stable argsort keeps equal scores in ascending-index order: include the
// k_rem smallest indices among items whose score == K-th value.
__global__ void __launch_bounds__(256)
k_equals(const float* __restrict__ model, const unsigned* __restrict__ scores_u,
         const SelState* __restrict__ st, float* __restrict__ eq_sum) {
  __shared__ unsigned idxs[EQ_CAP];
  __shared__ unsigned cnt;
  if (threadIdx.x == 0) cnt = 0u;
  __syncthreads();
  const unsigned Tu = st->prefix;
  for (int i = threadIdx.x; i < N_ANCH; i += 256) {
    if (scores_u[i] == Tu) {
      unsigned p = atomicAdd(&cnt, 1u);
      if (p < EQ_CAP) idxs[p] = (unsigned)i;
    }
  }
  __syncthreads();
  if (threadIdx.x == 0) {
    int n = (cnt < EQ_CAP) ? (int)cnt : EQ_CAP;
    for (int a = 1; a < n; ++a) {                  // ascending-index sort (n ~ 1)
      unsigned key = idxs[a];
      int b = a - 1;
      while (b >= 0 && idxs[b] > key) { idxs[b + 1] = idxs[b]; --b; }
      idxs[b + 1] = key;
    }
    int need = (int)st->k_rem;
    if (need > n) need = n;
    float s = 0.0f;
    if (Tu >= f2u(CONF_T)) {
      const float sc = u2f(Tu);
      for (int r = 0; r < need; ++r) {
        int i = (int)idxs[r];
        s += sc + model[0 * N_ANCH + i] + model[1 * N_ANCH + i]
                + model[2 * N_ANCH + i] + model[3 * N_ANCH + i];
      }
    }
    *eq_sum = s;
  }
}

// ---------------------------------------------------------------- combine
__global__ void k_combine(const float* __restrict__ partials, int nb,
                          const float* __restrict__ eq_sum,
                          float* __restrict__ out) {
  float s = *eq_sum;
  for (int i = 0; i < nb; ++i) s += partials[i];
  out[0] = s;
}

// ================================================================ launch
extern "C" void kernel_launch(void* const* d_in, const int* in_sizes, int n_in,
                              void* d_out, int out_size, void* d_ws, size_t ws_size,
                              hipStream_t stream) {
  (void)in_sizes; (void)n_in; (void)out_size; (void)ws_size;
  const float* model = (const float*)d_in[0];      // (84, 336000) row-major
  float* out = (float*)d_out;

  char* ws = (char*)d_ws;
  size_t off = 0;
  unsigned* scores_u = (unsigned*)(ws + off); off += (size_t)N_ANCH * sizeof(unsigned);
  unsigned* hist = (unsigned*)(ws + off);     off += 256 * sizeof(unsigned);
  SelState* st = (SelState*)(ws + off);       off += sizeof(SelState);
  off = (off + 15) & ~(size_t)15;
  float* eq_sum = (float*)(ws + off);         off += 16;
  float* partials = (float*)(ws + off);       off += NB_SUM * sizeof(float);

  k_init<<<1, 256, 0, stream>>>(hist, st, eq_sum);
  k_zero<<<(N_ANCH + 255) / 256, 256, 0, stream>>>(scores_u);
  dim3 sg((N_V4 + 255) / 256, ROW_BLOCKS);
  k_scores<<<sg, 256, 0, stream>>>(model, scores_u);
  for (int p = 0; p < 4; ++p) {
    int shift = 24 - 8 * p;
    k_hist<<<120, 256, 0, stream>>>(scores_u, hist, st, shift);
    k_select<<<1, 1, 0, stream>>>(hist, st, shift);
  }
  k_sum<<<NB_SUM, 256, 0, stream>>>(model, scores_u, st, partials);
  k_equals<<<1, 256, 0, stream>>>(model, scores_u, st, eq_sum);
  k_combine<<<1, 1, 0, stream>>>(partials, NB_SUM, eq_sum, out);
}